// SqueezeExcitation_3539053052006
// MI455X (gfx1250) — compile-verified
//
#include <hip/hip_runtime.h>
#include <hip/hip_bf16.h>

typedef __attribute__((ext_vector_type(2))) float v2f;
typedef __attribute__((ext_vector_type(8))) float v8f;

#define C_DIM 128
#define S_DIM 32
#define G_DIM 1024
#define SEG_ROWS 512

// ---------------- zero-init workspace region ----------------
__global__ void se_zero_kernel(float* p, int n) {
    int i = blockIdx.x * blockDim.x + threadIdx.x;
    if (i < n) p[i] = 0.0f;
}

// ---------------- segment sums + counts (batch is sorted) ----------------
// 128 threads = one per column. Each block handles SEG_ROWS consecutive rows.
// Row reads are fully coalesced (512B per row across the block). Sorted batch
// -> register run accumulation, one float atomic per (run, column).
__global__ void se_segsum_kernel(const float* __restrict__ x,
                                 const long long* __restrict__ batch,
                                 float* __restrict__ sums,
                                 float* __restrict__ counts) {
    __shared__ int sgid[SEG_ROWS];
    const int base = blockIdx.x * SEG_ROWS;
    const int c = threadIdx.x;  // 0..127

    for (int r = threadIdx.x; r < SEG_ROWS; r += blockDim.x)
        sgid[r] = (int)batch[base + r];
    __syncthreads();

    if (threadIdx.x == 0) {
        int cur = sgid[0];
        int run = 0;
        for (int r = 0; r < SEG_ROWS; ++r) {
            int g = sgid[r];
            if (g != cur) {
                atomicAdd(&counts[cur], (float)run);
                cur = g; run = 0;
            }
            ++run;
        }
        atomicAdd(&counts[cur], (float)run);
    }

    const float* xp = x + (long long)base * C_DIM + c;
    float acc = 0.0f;
    int cur = sgid[0];
    for (int r = 0; r < SEG_ROWS; ++r) {
        int g = sgid[r];
        if (g != cur) {
            atomicAdd(&sums[cur * C_DIM + c], acc);
            acc = 0.0f; cur = g;
        }
        acc += xp[(long long)r * C_DIM];
    }
    atomicAdd(&sums[cur * C_DIM + c], acc);
}

// ---------------- squeeze = relu(mean @ W_s^T + b_s) via WMMA f32 16x16x4 ----
// One wave (32 threads) per 16x16 output tile. grid = (G/16) * (S/16) = 128.
__global__ void se_squeeze_kernel(const float* __restrict__ sums,
                                  const float* __restrict__ counts,
                                  const float* __restrict__ W_s,
                                  const float* __restrict__ b_s,
                                  float* __restrict__ squeeze) {
    const int tile = blockIdx.x;
    const int tm = tile >> 1;       // 0..63  (graph tile)
    const int tn = tile & 1;        // 0..1   (S tile)
    const int lane = threadIdx.x;
    const int l15 = lane & 15;
    const int hi = lane >> 4;       // 0: K even base, 1: K+2

    const int m = tm * 16 + l15;    // A row for this lane
    const int n = tn * 16 + l15;    // B col for this lane
    const float inv = 1.0f / fmaxf(counts[m], 1.0f);

    v8f acc = {};
#pragma unroll
    for (int ks = 0; ks < C_DIM / 4; ++ks) {
        const int k0 = ks * 4 + hi * 2;
        v2f a, b;
        a.x = sums[m * C_DIM + k0] * inv;       // mean[m][k0]
        a.y = sums[m * C_DIM + k0 + 1] * inv;   // mean[m][k0+1]
        b.x = W_s[n * C_DIM + k0];              // B[k][n] = W_s[n][k]
        b.y = W_s[n * C_DIM + k0 + 1];
        acc = __builtin_amdgcn_wmma_f32_16x16x4_f32(
            false, a, false, b, (short)0, acc, false, false);
    }

#pragma unroll
    for (int r = 0; r < 8; ++r) {
        const int row = tm * 16 + r + hi * 8;   // C/D layout: vgpr r, lane half
        const int col = tn * 16 + l15;
        const float v = acc[r] + b_s[col];
        squeeze[row * S_DIM + col] = fmaxf(v, 0.0f);
    }
}

// ---------------- excite = sigmoid(squeeze @ W_e^T + b_e) via WMMA ----------
// grid = (G/16) * (C/16) = 64 * 8 = 512 waves.
__global__ void se_excite_kernel(const float* __restrict__ squeeze,
                                 const float* __restrict__ W_e,
                                 const float* __restrict__ b_e,
                                 float* __restrict__ excite) {
    const int tile = blockIdx.x;
    const int tm = tile >> 3;       // 0..63
    const int tn = tile & 7;        // 0..7
    const int lane = threadIdx.x;
    const int l15 = lane & 15;
    const int hi = lane >> 4;

    const int m = tm * 16 + l15;
    const int n = tn * 16 + l15;

    v8f acc = {};
#pragma unroll
    for (int ks = 0; ks < S_DIM / 4; ++ks) {
        const int k0 = ks * 4 + hi * 2;
        v2f a, b;
        a.x = squeeze[m * S_DIM + k0];
        a.y = squeeze[m * S_DIM + k0 + 1];
        b.x = W_e[n * S_DIM + k0];              // B[k][n] = W_e[n][k]
        b.y = W_e[n * S_DIM + k0 + 1];
        acc = __builtin_amdgcn_wmma_f32_16x16x4_f32(
            false, a, false, b, (short)0, acc, false, false);
    }

#pragma unroll
    for (int r = 0; r < 8; ++r) {
        const int row = tm * 16 + r + hi * 8;
        const int col = tn * 16 + l15;
        const float v = acc[r] + b_e[col];
        excite[row * C_DIM + col] = 1.0f / (1.0f + __expf(-v));
    }
}

// ---------------- out = x * excite[batch], float4 vectorized ----------------
__global__ void se_scale_kernel(const float* __restrict__ x,
                                const long long* __restrict__ batch,
                                const float* __restrict__ excite,
                                float* __restrict__ out) {
    const long long tid = (long long)blockIdx.x * blockDim.x + threadIdx.x;
    const int row = (int)(tid >> 5);        // C/4 = 32 quads per row
    const int q = (int)(tid & 31);
    const int g = (int)batch[row];

    const float4 xv = ((const float4*)x)[tid];
    const float4 ev = ((const float4*)(excite + (long long)g * C_DIM))[q];
    float4 o;
    o.x = xv.x * ev.x;
    o.y = xv.y * ev.y;
    o.z = xv.z * ev.z;
    o.w = xv.w * ev.w;
    ((float4*)out)[tid] = o;
}

extern "C" void kernel_launch(void* const* d_in, const int* in_sizes, int n_in,
                              void* d_out, int out_size, void* d_ws, size_t ws_size,
                              hipStream_t stream) {
    const float*      x     = (const float*)d_in[0];
    const long long*  batch = (const long long*)d_in[1];
    const float*      W_s   = (const float*)d_in[2];
    const float*      b_s   = (const float*)d_in[3];
    const float*      W_e   = (const float*)d_in[4];
    const float*      b_e   = (const float*)d_in[5];
    float*            out   = (float*)d_out;

    const int N = in_sizes[1];              // number of nodes (batch length)

    // workspace layout (floats)
    float* ws      = (float*)d_ws;
    float* sums    = ws;                          // G*C     = 131072
    float* counts  = ws + G_DIM * C_DIM;          // G       = 1024
    float* squeeze = counts + G_DIM;              // G*S     = 32768
    float* excite  = squeeze + G_DIM * S_DIM;     // G*C     = 131072

    const int zero_n = G_DIM * C_DIM + G_DIM;     // sums + counts
    se_zero_kernel<<<(zero_n + 255) / 256, 256, 0, stream>>>(ws, zero_n);

    se_segsum_kernel<<<N / SEG_ROWS, C_DIM, 0, stream>>>(x, batch, sums, counts);

    se_squeeze_kernel<<<(G_DIM / 16) * (S_DIM / 16), 32, 0, stream>>>(
        sums, counts, W_s, b_s, squeeze);

    se_excite_kernel<<<(G_DIM / 16) * (C_DIM / 16), 32, 0, stream>>>(
        squeeze, W_e, b_e, excite);

    // N*C/4 float4 elements, 256 threads/block
    const long long quads = (long long)N * C_DIM / 4;
    se_scale_kernel<<<(unsigned)(quads / 256), 256, 0, stream>>>(
        x, batch, excite, out);
}